// ConvAttentionModule_65481071397669
// MI455X (gfx1250) — compile-verified
//
#include <hip/hip_runtime.h>

// ConvAttention collapses to:
//   s[n]  = tanh( (1/32) * sum_{b,2x2patch} x )          n in [0,4096)
//   Z[n]  = sum_m exp(8 s[n] s[m]);  Y[n] = sum_m exp(8 s[n] s[m]) s[m]
//   out[b]= (1/4096) * sum_n Y[n]/Z[n] * sum_d proj_w[d] + proj_b   (all b equal)

#define N_PATCH 4096     // 64*64 patch positions
#define DIMX    65
#define BATCH   8
#define SCALE   8.0f     // D / sqrt(D) = 64 / 8

typedef float v2f __attribute__((ext_vector_type(2)));
typedef float v8f __attribute__((ext_vector_type(8)));

// ---------------- Kernel A: patch scalars s[n] ----------------
__global__ void k_scalar(const float* __restrict__ x, float* __restrict__ s) {
  int n = blockIdx.x * blockDim.x + threadIdx.x;   // exactly 4096 threads
  int i = n >> 6, j = n & 63;
  const float* p = x + i * DIMX + j;
  float sum = 0.0f;
  #pragma unroll
  for (int b = 0; b < BATCH; ++b) {
    const float* q = p + b * (DIMX * DIMX);
    sum += q[0] + q[1] + q[DIMX] + q[DIMX + 1];
  }
  s[n] = tanhf(sum * (1.0f / 32.0f));
}

// ------- Kernel B: softmax row sums via f32 WMMA reductions -------
// Block handles 16 rows (n-tile). Each wave owns m-columns {wave*4 + 32k}.
// A tile = exp(8 s[n] s[m]) (16x4 f32), B[:,0]=1, B[:,1]=s[m] -> D col0=Z, col1=Y.
__global__ void __launch_bounds__(256) k_attn(const float* __restrict__ s,
                                              float* __restrict__ partial) {
  __shared__ float sh_s[N_PATCH];   // 16KB of 320KB/WGP
  __shared__ float Zp[8][16];
  __shared__ float Yp[8][16];
  __shared__ float rowval[16];

  for (int t = threadIdx.x; t < N_PATCH; t += 256) sh_s[t] = s[t];  // uniform: 4096%256==0
  __syncthreads();

  const int lane = threadIdx.x & 31;
  const int wave = threadIdx.x >> 5;
  const int col  = lane & 15;                 // A: row M; B: column N; D: column N
  const int k0   = (lane >> 4) << 1;          // K pair: 0 or 2 (per 16x4 f32 A layout)
  const int nb   = blockIdx.x * 16;
  const float sn = sh_s[nb + col] * SCALE;    // 8 * s[row], row = lane%16

  v8f c = {};
  int idx = wave * 4 + k0;                    // global m for this lane's K pair
  #pragma unroll 2
  for (int it = 0; it < N_PATCH / 32; ++it, idx += 32) {
    float sm0 = sh_s[idx];
    float sm1 = sh_s[idx + 1];
    v2f a, b;
    a.x = __expf(sn * sm0);                   // E[row, m+k0]
    a.y = __expf(sn * sm1);                   // E[row, m+k0+1]
    b.x = (col == 0) ? 1.0f : ((col == 1) ? sm0 : 0.0f);   // B[k0,  col]
    b.y = (col == 0) ? 1.0f : ((col == 1) ? sm1 : 0.0f);   // B[k0+1,col]
    // D += A x B : col0 accumulates Z partials, col1 accumulates Y partials
    c = __builtin_amdgcn_wmma_f32_16x16x4_f32(false, a, false, b,
                                              (short)0, c, false, false);
  }

  // D layout: VGPR r -> row (lane<16 ? r : 8+r), column = lane%16
  const int rbase = (lane < 16) ? 0 : 8;
  if (col == 0) {
    #pragma unroll
    for (int r = 0; r < 8; ++r) Zp[wave][rbase + r] = c[r];
  } else if (col == 1) {
    #pragma unroll
    for (int r = 0; r < 8; ++r) Yp[wave][rbase + r] = c[r];
  }
  __syncthreads();

  if (threadIdx.x < 16) {                     // deterministic cross-wave sum
    float z = 0.0f, y = 0.0f;
    #pragma unroll
    for (int w = 0; w < 8; ++w) { z += Zp[w][threadIdx.x]; y += Yp[w][threadIdx.x]; }
    rowval[threadIdx.x] = y / z;              // ctx value for this row
  }
  __syncthreads();
  if (threadIdx.x == 0) {
    float t = 0.0f;
    #pragma unroll
    for (int r = 0; r < 16; ++r) t += rowval[r];
    partial[blockIdx.x] = t;                  // one deterministic partial per block
  }
}

// ---------------- Kernel C: final projection ----------------
__global__ void k_final(const float* __restrict__ w, const float* __restrict__ bias,
                        const float* __restrict__ partial, float* __restrict__ out) {
  int lane = threadIdx.x;                     // single wave32
  float acc = 0.0f;
  #pragma unroll
  for (int k = 0; k < 8; ++k) acc += partial[lane + 32 * k];   // 256 partials
  float wv = w[lane] + w[lane + 32];                           // sum_d proj_w
  #pragma unroll
  for (int off = 16; off > 0; off >>= 1) {
    acc += __shfl_down(acc, off, 32);
    wv  += __shfl_down(wv,  off, 32);
  }
  if (lane == 0) {
    float res = (acc * (1.0f / (float)N_PATCH)) * wv + bias[0];
    #pragma unroll
    for (int b = 0; b < BATCH; ++b) out[b] = res;              // 8 identical outputs
  }
}

extern "C" void kernel_launch(void* const* d_in, const int* in_sizes, int n_in,
                              void* d_out, int out_size, void* d_ws, size_t ws_size,
                              hipStream_t stream) {
  const float* x  = (const float*)d_in[0];   // (8,1,65,65) f32
  const float* pw = (const float*)d_in[1];   // (1,64) f32
  const float* pb = (const float*)d_in[2];   // (1,)  f32
  float* out      = (float*)d_out;           // 8 f32
  float* s        = (float*)d_ws;            // 4096 f32
  float* partial  = s + N_PATCH;             // 256 f32

  k_scalar<<<N_PATCH / 256, 256, 0, stream>>>(x, s);
  k_attn  <<<N_PATCH / 16,  256, 0, stream>>>(s, partial);
  k_final <<<1, 32, 0, stream>>>(pw, pb, partial, out);

  (void)in_sizes; (void)n_in; (void)out_size; (void)ws_size;
}